// SingleCoreAttention_2319282340148
// MI455X (gfx1250) — compile-verified
//
#include <hip/hip_runtime.h>
#include <hip/hip_bf16.h>

typedef __attribute__((ext_vector_type(2))) float v2f;
typedef __attribute__((ext_vector_type(8))) float v8f;

#define TEMPERATURE 8.0f

// ---------------------------------------------------------------------------
// Kernel 1: coef[b][t] = 0.5*(vec[0][t]+vec[1][t])/T * sum_j k[b][j][t]
// One block per batch, 64 threads (t). Loads coalesced across t.
// ---------------------------------------------------------------------------
__global__ __launch_bounds__(64) void coef_kernel(const float* __restrict__ k,
                                                  const float* __restrict__ vec,
                                                  float* __restrict__ coef,
                                                  int L) {
    const int b = blockIdx.x;
    const int t = threadIdx.x;                       // 0..63
    const float* kb = k + (size_t)b * L * 64 + t;
    float s = 0.0f;
    for (int j = 0; j < L; ++j) s += kb[(size_t)j * 64];
    const float w = 0.5f * (vec[t] + vec[64 + t]) * (1.0f / TEMPERATURE);
    coef[b * 64 + t] = w * s;
}

// ---------------------------------------------------------------------------
// Kernel 2: out[b][i][r] = sum_t q[b][i][t] * coef[b][t] * v[b][t][r]
// Block handles 4 rows of one batch; W = diag(coef)*V staged in LDS.
// ---------------------------------------------------------------------------
__global__ __launch_bounds__(256) void out_kernel(const float* __restrict__ q,
                                                  const float* __restrict__ v,
                                                  const float* __restrict__ coef,
                                                  float* __restrict__ out,
                                                  int L) {
    __shared__ float W[64 * 64];
    __shared__ float qs[4 * 64];
    const int b  = blockIdx.y;
    const int i0 = blockIdx.x * 4;

    for (int idx = threadIdx.x; idx < 64 * 64; idx += 256) {
        const int t = idx >> 6;
        const int r = idx & 63;
        W[idx] = coef[b * 64 + t] * v[((size_t)b * L + t) * 64 + r];
    }
    for (int idx = threadIdx.x; idx < 4 * 64; idx += 256) {
        const int row = idx >> 6;
        qs[idx] = q[((size_t)b * L + i0 + row) * 64 + (idx & 63)];
    }
    __syncthreads();

    const int r   = threadIdx.x & 63;
    const int row = threadIdx.x >> 6;                // 0..3
    float acc = 0.0f;
#pragma unroll
    for (int t = 0; t < 64; ++t)
        acc += qs[row * 64 + t] * W[t * 64 + r];
    out[((size_t)b * L + i0 + row) * 64 + r] = acc;
}

// ---------------------------------------------------------------------------
// Kernel 3: attn[b][i][j] = sum_d q[b][i][d] * k[b][j][d]
// One wave per 16x16 tile; 4 waves/block cover 16 rows x 64 cols.
// Uses V_WMMA_F32_16X16X4_F32, K=64 via 16 chained WMMAs.
//
// fp32 WMMA A-layout (16x4, 2 VGPRs/lane):
//   VGPR0: lanes 0-15 -> (M=l, K=4s+0), lanes 16-31 -> (M=l-16, K=4s+2)
//   VGPR1: same M, K+1  => lane l reads float2 at row (l&15), col 4s+2*(l>>4)
// B (4x16) mirrors this with N=(l&15), K rows from k[b][j0+N][d] — identical
// per-lane addressing with k instead of q.
// D (16x16, 8 VGPRs): VGPR r -> row i0 + r + 8*(l>>4), col j0 + (l&15).
// ---------------------------------------------------------------------------
__global__ __launch_bounds__(128) void attn_kernel(const float* __restrict__ q,
                                                   const float* __restrict__ k,
                                                   float* __restrict__ attn,
                                                   int L) {
    const int lane = threadIdx.x & 31;
    const int wave = threadIdx.x >> 5;               // 0..3
    const int b    = blockIdx.z;
    const int i0   = blockIdx.y * 16;                // row tile
    const int j0   = (blockIdx.x * 4 + wave) * 16;   // col tile
    const int half = lane >> 4;                      // 0 or 1
    const int l15  = lane & 15;

    const float* qrow = q + ((size_t)b * L + i0 + l15) * 64 + 2 * half;
    const float* krow = k + ((size_t)b * L + j0 + l15) * 64 + 2 * half;

    v8f c = {};
#pragma unroll
    for (int s = 0; s < 16; ++s) {
        v2f a  = *(const v2f*)(qrow + 4 * s);
        v2f bm = *(const v2f*)(krow + 4 * s);
        c = __builtin_amdgcn_wmma_f32_16x16x4_f32(
                /*neg_a=*/false, a, /*neg_b=*/false, bm,
                /*c_mod=*/(short)0, c, /*reuse_a=*/false, /*reuse_b=*/false);
    }

    float* outp = attn + ((size_t)b * L + i0 + 8 * half) * L + j0 + l15;
#pragma unroll
    for (int r = 0; r < 8; ++r)
        outp[(size_t)r * L] = c[r];
}

// ---------------------------------------------------------------------------
extern "C" void kernel_launch(void* const* d_in, const int* in_sizes, int n_in,
                              void* d_out, int out_size, void* d_ws, size_t ws_size,
                              hipStream_t stream) {
    const float* q   = (const float*)d_in[0];
    const float* k   = (const float*)d_in[1];
    const float* v   = (const float*)d_in[2];
    const float* vec = (const float*)d_in[3];

    const int B = 4;
    const int D = 64;
    const int L = in_sizes[0] / (B * D);             // 4096

    float* out  = (float*)d_out;                     // [B, L, D]
    float* attn = out + (size_t)B * L * D;           // [B, L, L]
    float* coef = (float*)d_ws;                      // [B, 64]

    coef_kernel<<<dim3(B), 64, 0, stream>>>(k, vec, coef, L);
    out_kernel<<<dim3(L / 4, B), 256, 0, stream>>>(q, v, coef, out, L);
    attn_kernel<<<dim3(L / 64, L / 16, B), 128, 0, stream>>>(q, k, attn, L);
}